// MemoryTree_90812788506712
// MI455X (gfx1250) — compile-verified
//
#include <hip/hip_runtime.h>

typedef __attribute__((ext_vector_type(2))) float v2f;
typedef __attribute__((ext_vector_type(8))) float v8f;

#define B_    8
#define LK    4
#define D_    64
#define L_    4096
#define DEPTH 12

// ---- optional CDNA5 async global->LDS path (guarded; falls back cleanly) ----
#if defined(__AMDGCN__) && __has_builtin(__builtin_amdgcn_global_load_async_to_lds_b128)
#define MT_ASYNC 1
typedef int mt_b128 __attribute__((vector_size(16)));           // 16-byte payload
typedef __attribute__((address_space(1))) mt_b128 mt_gb128;     // global
typedef __attribute__((address_space(3))) mt_b128 mt_lb128;     // LDS

__device__ __forceinline__ void mt_wait_async8() {
#if __has_builtin(__builtin_amdgcn_s_wait_asynccnt)
    __builtin_amdgcn_s_wait_asynccnt(8);
#else
    asm volatile("s_wait_asynccnt 0x8" ::: "memory");
#endif
    asm volatile("" ::: "memory");   // keep LDS reads after the wait
}
__device__ __forceinline__ void mt_wait_async0() {
#if __has_builtin(__builtin_amdgcn_s_wait_asynccnt)
    __builtin_amdgcn_s_wait_asynccnt(0);
#else
    asm volatile("s_wait_asynccnt 0x0" ::: "memory");
#endif
    asm volatile("" ::: "memory");
}
#endif

// ---------------------------------------------------------------------------
// Stage 1: leaf logits r[b,n,k] = (q[b,k]^T . leafs[b,n] . v[b,k]) / 64
// GEMV form: R = Mflat(rows = vec(leaf matrices)) @ Qflat,
// Qflat[j,k] = q[b,k][j>>6] * v[b,k][j&63], padded to 16 cols (4..15 = 0).
// One wave per 16-row tile; V_WMMA_F32_16X16X4_F32 accumulates K=4096.
// Async path: double-buffered 4KB A-chunks streamed global->LDS (ASYNCcnt).
// ---------------------------------------------------------------------------
__global__ __launch_bounds__(32) void mt_leaf_logits_kernel(
    const float* __restrict__ q, const float* __restrict__ v,
    const float* __restrict__ leafs, float* __restrict__ rws)
{
    __shared__ __align__(16) float lq[16 * 64];
    __shared__ __align__(16) float lv[16 * 64];
#ifdef MT_ASYNC
    __shared__ __align__(16) float abuf[2][16 * 64];   // 2 x 4KB A-chunk buffers
#endif

    const int lane = threadIdx.x;       // 0..31 (wave32)
    const int b    = blockIdx.x >> 8;   // 256 tiles per batch
    const int tile = blockIdx.x & 255;

    // zero columns 4..15 so padded B-columns contribute exact zeros
    for (int i = lane; i < 16 * 64; i += 32) { lq[i] = 0.0f; lv[i] = 0.0f; }
    __syncthreads();
    for (int i = lane; i < LK * 64; i += 32) {
        lq[i] = q[(size_t)b * LK * 64 + i];   // lq[k*64 + d]
        lv[i] = v[(size_t)b * LK * 64 + i];   // lv[k*64 + e]
    }
    __syncthreads();

    // WMMA f32 16x16x4 operand mapping (wave32):
    //  A (16x4): lane = {half,pos} -> row=pos, K = half*2 + {0,1}
    //  B (4x16): lane = {half,pos} -> col=pos, K = half*2 + {0,1}
    const int half = lane >> 4;
    const int pos  = lane & 15;
    const int k0   = half * 2;

    v8f acc = {0.f, 0.f, 0.f, 0.f, 0.f, 0.f, 0.f, 0.f};

#ifdef MT_ASYNC
    // Async streaming: chunk d = 16 rows x 64 f32 (rows contiguous 256B each).
    // Issue i covers rows 2i,2i+1: lanes 0..15 -> row 2i segs 0..15 (contiguous),
    // lanes 16..31 -> row 2i+1. 8 issues per 4KB chunk.
    const int seg = lane & 15;
    const float* gbase = leafs + (size_t)(b * L_ + tile * 16) * 4096;

    #pragma unroll
    for (int i = 0; i < 8; ++i) {                 // prime chunk 0
        const int row = 2 * i + half;
        const float* g = gbase + (size_t)row * 4096 + seg * 4;
        __builtin_amdgcn_global_load_async_to_lds_b128(
            (mt_gb128*)g, (mt_lb128*)&abuf[0][row * 64 + seg * 4], 0, 0);
    }

    for (int d = 0; d < 64; ++d) {
        const int sel = d & 1;
        if (d < 63) {
            #pragma unroll
            for (int i = 0; i < 8; ++i) {         // stream next chunk
                const int row = 2 * i + half;
                const float* g = gbase + (size_t)row * 4096 + (d + 1) * 64 + seg * 4;
                __builtin_amdgcn_global_load_async_to_lds_b128(
                    (mt_gb128*)g, (mt_lb128*)&abuf[sel ^ 1][row * 64 + seg * 4], 0, 0);
            }
            mt_wait_async8();                     // chunk d landed (in-order)
        } else {
            mt_wait_async0();
        }
        const float qd = lq[pos * 64 + d];
        #pragma unroll 4
        for (int t = 0; t < 16; ++t) {
            v2f a = *(const v2f*)(&abuf[sel][pos * 64 + t * 4 + k0]);   // ds_load_b64
            float2 vv = *(const float2*)(&lv[pos * 64 + t * 4 + k0]);  // ds_load_b64
            v2f bf;
            bf.x = qd * vv.x;
            bf.y = qd * vv.y;
            acc = __builtin_amdgcn_wmma_f32_16x16x4_f32(
                false, a, false, bf, (short)0, acc, false, false);
        }
    }
#else
    // Direct-load path (+ global_prefetch_b8 of each row's next 256B chunk)
    const float* __restrict__ Arow =
        leafs + ((size_t)(b * L_ + tile * 16 + pos)) * 4096 + k0;

    for (int d = 0; d < 64; ++d) {
        const float  qd = lq[pos * 64 + d];
        const float* Ap = Arow + d * 64;
        if (d < 63) __builtin_prefetch(Ap + 64, 0, 0);   // global_prefetch_b8
        #pragma unroll 4
        for (int t = 0; t < 16; ++t) {
            v2f a = *(const v2f*)(Ap + t * 4);                        // global_load_b64
            float2 vv = *(const float2*)(&lv[pos * 64 + t * 4 + k0]); // ds_load_b64
            v2f bf;
            bf.x = qd * vv.x;
            bf.y = qd * vv.y;
            acc = __builtin_amdgcn_wmma_f32_16x16x4_f32(
                false, a, false, bf, (short)0, acc, false, false);
        }
    }
#endif

    // C layout: lane {half,pos}, VGPR g -> row = half*8 + g, col = pos
    if (pos < LK) {
        const float scale = 1.0f / (float)D_;
        const int nbase = tile * 16 + half * 8;
        #pragma unroll
        for (int g = 0; g < 8; ++g) {
            rws[((size_t)(b * L_ + nbase + g)) * LK + pos] = acc[g] * scale;
        }
    }
}

// ---------------------------------------------------------------------------
// Stage 2: per (b,k): pairwise-pool leaf logits through 12 levels; at each
// level compute nll = logsumexp - logit[label] and unnormalized weight
// wy = 32 / (count(label)+1e-8).  (Global w-normalization cancels in per_q.)
// ---------------------------------------------------------------------------
__global__ __launch_bounds__(256) void mt_levels_kernel(
    const float* __restrict__ rws, const int* __restrict__ expected,
    float* __restrict__ out2 /* [DEPTH][32][2] = {nll, wy} */)
{
    __shared__ float buf[L_];     // 16 KB: current-level logits
    __shared__ float red[256];
    __shared__ int   labs[B_ * LK];

    const int tid = threadIdx.x;
    const int b   = blockIdx.x >> 2;
    const int k   = blockIdx.x & 3;

    if (tid < B_ * LK) labs[tid] = expected[tid];
    for (int n = tid; n < L_; n += 256)
        buf[n] = rws[((size_t)(b * L_ + n)) * LK + k];
    __syncthreads();

    const int mylab = labs[b * LK + k];

    int ncls = L_;
    for (int level = 0; level < DEPTH; ++level) {
        // --- logsumexp over buf[0..ncls) ---
        float lmax = -3.402823466e38f;
        for (int n = tid; n < ncls; n += 256) lmax = fmaxf(lmax, buf[n]);
        red[tid] = lmax; __syncthreads();
        for (int s = 128; s > 0; s >>= 1) {
            if (tid < s) red[tid] = fmaxf(red[tid], red[tid + s]);
            __syncthreads();
        }
        const float gmax = red[0]; __syncthreads();

        float lsum = 0.0f;
        for (int n = tid; n < ncls; n += 256) lsum += expf(buf[n] - gmax);
        red[tid] = lsum; __syncthreads();
        for (int s = 128; s > 0; s >>= 1) {
            if (tid < s) red[tid] += red[tid + s];
            __syncthreads();
        }
        const float lse = gmax + logf(red[0]);
        __syncthreads();

        if (tid == 0) {
            const int lab = mylab >> level;
            const float nll = lse - buf[lab];
            int cnt = 0;
            #pragma unroll
            for (int i = 0; i < B_ * LK; ++i)
                cnt += ((labs[i] >> level) == lab) ? 1 : 0;
            const float wy = 32.0f / ((float)cnt + 1e-8f);
            out2[(level * 32 + b * LK + k) * 2 + 0] = nll;
            out2[(level * 32 + b * LK + k) * 2 + 1] = wy;
        }
        __syncthreads();

        // --- pool to next level: buf[n] = 0.5*(buf[2n]+buf[2n+1]) ---
        const int next = ncls >> 1;
        float tmp[8];                      // next <= 2048 -> <= 8 per thread
        int c = 0;
        for (int n = tid; n < next; n += 256)
            tmp[c++] = 0.5f * (buf[2 * n] + buf[2 * n + 1]);
        __syncthreads();
        c = 0;
        for (int n = tid; n < next; n += 256) buf[n] = tmp[c++];
        __syncthreads();
        ncls = next;
    }
}

// ---------------------------------------------------------------------------
// Stage 3: total = sum over (level,k) of  sum_b(wy*nll) / sum_b(wy)
// ---------------------------------------------------------------------------
__global__ __launch_bounds__(64) void mt_final_kernel(
    const float* __restrict__ out2, float* __restrict__ out)
{
    __shared__ float red[64];
    const int tid = threadIdx.x;
    float term = 0.0f;
    if (tid < DEPTH * LK) {
        const int level = tid >> 2;
        const int k     = tid & 3;
        float num = 0.0f, den = 0.0f;
        #pragma unroll
        for (int b = 0; b < B_; ++b) {
            const float nll = out2[(level * 32 + b * LK + k) * 2 + 0];
            const float wy  = out2[(level * 32 + b * LK + k) * 2 + 1];
            num += wy * nll;
            den += wy;
        }
        term = num / den;
    }
    red[tid] = term; __syncthreads();
    for (int s = 32; s > 0; s >>= 1) {
        if (tid < s) red[tid] += red[tid + s];
        __syncthreads();
    }
    if (tid == 0) out[0] = red[0];
}

// ---------------------------------------------------------------------------
extern "C" void kernel_launch(void* const* d_in, const int* in_sizes, int n_in,
                              void* d_out, int out_size, void* d_ws, size_t ws_size,
                              hipStream_t stream)
{
    const float* q        = (const float*)d_in[0];
    const float* v        = (const float*)d_in[1];
    const int*   expected = (const int*)d_in[2];
    const float* leafs    = (const float*)d_in[3];
    float* out = (float*)d_out;

    float* rws = (float*)d_ws;                       // 8*4096*4 f32 = 512 KB
    float* ws2 = rws + (size_t)B_ * L_ * LK;         // DEPTH*32*2 f32

    mt_leaf_logits_kernel<<<B_ * (L_ / 16), 32, 0, stream>>>(q, v, leafs, rws);
    mt_levels_kernel<<<B_ * LK, 256, 0, stream>>>(rws, expected, ws2);
    mt_final_kernel<<<1, 64, 0, stream>>>(ws2, out);
}